// CreateMultiScaleWarping_56135222558988
// MI455X (gfx1250) — compile-verified
//
#include <hip/hip_runtime.h>

typedef float v4f __attribute__((ext_vector_type(4)));

constexpr int ilog2(int v) { int s = 0; while ((1 << s) < v) ++s; return s; }

// Multi-scale bilinear warp (zero-padded), NCHW, H == W == power of two.
// One thread = 4 consecutive x pixels for one (b, y); all 3 channels.
// W is a template parameter: strides, masks and (W-1)/W are immediates.
template <int W>
__global__ __launch_bounds__(256) void warp_bilinear_kernel(
    const float* __restrict__ ref,   // [B, 3, W, W]
    const float* __restrict__ flow,  // [B, 2, W, W]  plane0 = fy, plane1 = fx
    float* __restrict__ out)         // [B, 3, W, W]
{
    constexpr int HW     = W * W;
    constexpr int QPR    = W >> 2;          // quads per row
    constexpr int QSHIFT = ilog2(QPR);
    constexpr float SC   = (float)(W - 1) / (float)W;  // (W-1)/H, H==W

    const int b = blockIdx.y;
    const int i = blockIdx.x * blockDim.x + threadIdx.x;  // quad index in image
    const int y = i >> QSHIFT;
    const int x = (i & (QPR - 1)) << 2;
    if (y >= W) return;  // grids are exact; kept for safety, costs one s_cmp

    const float* __restrict__ refb  = ref  + (size_t)b * 3 * HW;
    const float* __restrict__ flowp = flow + (size_t)b * 2 * HW;

    const int rowoff = y * W + x;  // 32-bit; plane <= 1 MB

    // Prefetch the destination neighborhood of ref for all 3 channels
    // (flow offsets are ~N(0,1) px, so gathers land near (y, x..x+3)).
    // global_prefetch_b8 on gfx1250; overlaps the flow-load latency.
    __builtin_prefetch(refb +          rowoff, 0, 3);
    __builtin_prefetch(refb +     HW + rowoff, 0, 3);
    __builtin_prefetch(refb + 2 * HW + rowoff, 0, 3);

    // Flow is read exactly once -> non-temporal loads preserve L2 for ref.
    const v4f fy4 = __builtin_nontemporal_load((const v4f*)(flowp + rowoff));
    const v4f fx4 = __builtin_nontemporal_load((const v4f*)(flowp + HW + rowoff));

    float acc[3][4];

#pragma unroll
    for (int j = 0; j < 4; ++j) {
        const float xf = (float)(x + j) + fx4[j] * SC;
        const float yf = (float)y       + fy4[j] * SC;

        const float x0f = __builtin_floorf(xf);
        const float y0f = __builtin_floorf(yf);
        const float tx  = xf - x0f;
        const float ty  = yf - y0f;

        const int ix0 = (int)x0f, iy0 = (int)y0f;
        const int ix1 = ix0 + 1,  iy1 = iy0 + 1;

        const bool vx0 = (ix0 >= 0) & (ix0 < W);
        const bool vx1 = (ix1 >= 0) & (ix1 < W);
        const bool vy0 = (iy0 >= 0) & (iy0 < W);
        const bool vy1 = (iy1 >= 0) & (iy1 < W);

        // Zero-padding folded into weights (matches ref: clamp index, mask value).
        const float wa = (1.f - tx) * (1.f - ty) * ((vx0 & vy0) ? 1.f : 0.f); // (x0,y0)
        const float wb = (1.f - tx) * ty         * ((vx0 & vy1) ? 1.f : 0.f); // (x0,y1)
        const float wc = tx         * (1.f - ty) * ((vx1 & vy0) ? 1.f : 0.f); // (x1,y0)
        const float wd = tx         * ty         * ((vx1 & vy1) ? 1.f : 0.f); // (x1,y1)

        const int cx0 = min(max(ix0, 0), W - 1);
        const int cx1 = min(max(ix1, 0), W - 1);
        const int cy0 = min(max(iy0, 0), W - 1);
        const int cy1 = min(max(iy1, 0), W - 1);

        const int o00 = cy0 * W + cx0;   // 32-bit gather offsets
        const int o01 = cy0 * W + cx1;
        const int o10 = cy1 * W + cx0;
        const int o11 = cy1 * W + cx1;

#pragma unroll
        for (int c = 0; c < 3; ++c) {
            const float* __restrict__ rc = refb + c * HW;
            acc[c][j] = rc[o00] * wa + rc[o10] * wb + rc[o01] * wc + rc[o11] * wd;
        }
    }

    // Output is write-once, never re-read -> non-temporal b128 stores.
    float* __restrict__ outb = out + (size_t)b * 3 * HW + rowoff;
#pragma unroll
    for (int c = 0; c < 3; ++c) {
        v4f v = {acc[c][0], acc[c][1], acc[c][2], acc[c][3]};
        __builtin_nontemporal_store(v, (v4f*)(outb + c * HW));
    }
}

template <int W>
static inline void launch_scale(const float* ref, const float* flow, float* out,
                                int B, hipStream_t stream) {
    constexpr int quads_per_img = W * (W >> 2);
    dim3 grid(quads_per_img / 256, B);
    warp_bilinear_kernel<W><<<grid, 256, 0, stream>>>(ref, flow, out);
}

extern "C" void kernel_launch(void* const* d_in, const int* in_sizes, int n_in,
                              void* d_out, int out_size, void* d_ws, size_t ws_size,
                              hipStream_t stream) {
    (void)in_sizes; (void)n_in; (void)out_size; (void)d_ws; (void)ws_size;

    static const int kScales[4] = {512, 256, 128, 64};
    const int B = 16;

    size_t out_off = 0;
    for (int s = 0; s < 4; ++s) {
        const int W = kScales[s];
        const float* ref  = (const float*)d_in[2 * s];
        const float* flow = (const float*)d_in[2 * s + 1];
        float* out = (float*)d_out + out_off;

        switch (W) {
            case 512: launch_scale<512>(ref, flow, out, B, stream); break;
            case 256: launch_scale<256>(ref, flow, out, B, stream); break;
            case 128: launch_scale<128>(ref, flow, out, B, stream); break;
            default:  launch_scale< 64>(ref, flow, out, B, stream); break;
        }
        out_off += (size_t)B * 3 * W * W;
    }
}